// SimpleAttentionLayer_39608188403712
// MI455X (gfx1250) — compile-verified
//
#include <hip/hip_runtime.h>

#define S_LEN 64
#define B_DIM 32
#define N_DIM 325
#define H_DIM 64
#define TILE_N 16

typedef float v2f __attribute__((ext_vector_type(2)));
typedef float v4f __attribute__((ext_vector_type(4)));
typedef float v8f __attribute__((ext_vector_type(8)));

__device__ __forceinline__ float fast_tanh(float x) {
    // tanh(x) = (e^{2x}-1)/(e^{2x}+1), clamped so e^{2x} stays finite
    x = fminf(fmaxf(x, -15.f), 15.f);
    float e = __expf(2.f * x);
    return (e - 1.f) / (e + 1.f);
}

__global__ __launch_bounds__(256, 1)
void attn_wmma_kernel(const float* __restrict__ enc,   // (S,B,N,H)
                      const float* __restrict__ dec,   // (B,N,H)
                      const float* __restrict__ W,     // (H,H) row-major W[kout][h]
                      const float* __restrict__ bias,  // (H)
                      float* __restrict__ out)         // (B,N,S)
{
    __shared__ float W_lds[H_DIM * H_DIM];          // 16 KB
    __shared__ float b_lds[H_DIM];
    __shared__ float dec_lds[TILE_N * H_DIM];       // 4 KB
    __shared__ float stage[2][TILE_N * H_DIM];      // 8 KB, one enc tile per s-parity group
    __shared__ float e_lds[S_LEN][TILE_N];          // 4 KB
    __shared__ float red_lds[2][4][TILE_N];         // cross-wave partials

    const int tid   = threadIdx.x;
    const int wave  = tid >> 5;
    const int lane  = tid & 31;
    const int half  = lane >> 4;     // which 16-lane half of the wave
    const int l15   = lane & 15;
    const int grp   = wave >> 2;     // s-parity group: waves 0-3 -> even s, 4-7 -> odd s
    const int otile = wave & 3;      // 16-wide output-column tile of the GEMM

    const int b  = blockIdx.y;
    const int n0 = blockIdx.x * TILE_N;

    // ---- cooperative loads: W, bias, dec tile ----
    {
        const v4f* W4 = (const v4f*)W;
        v4f* Wl4 = (v4f*)W_lds;
        #pragma unroll
        for (int q = 0; q < 4; ++q)
            Wl4[tid + 256 * q] = W4[tid + 256 * q];
        if (tid < H_DIM) b_lds[tid] = bias[tid];
        int r = tid >> 4, h4 = tid & 15;
        int n = n0 + r; if (n > N_DIM - 1) n = N_DIM - 1;   // clamp (rows never written)
        ((v4f*)dec_lds)[tid] =
            ((const v4f*)dec)[(size_t)(b * N_DIM + n) * (H_DIM / 4) + h4];
    }
    __syncthreads();

    // ---- preload this wave's B fragments (held for all 64 s) ----
    // GEMM B element (K=h, Nc=kout) = W[kout][h]; VGPR j holds K = 4*kk + j + 2*half
    v2f bf[16];
    #pragma unroll
    for (int kk = 0; kk < 16; ++kk) {
        int addr = (16 * otile + l15) * H_DIM + 4 * kk + 2 * half;
        bf[kk].x = W_lds[addr];
        bf[kk].y = W_lds[addr + 1];
    }
    const float bias_val = b_lds[16 * otile + l15];

    // ---- loop over sequence positions (2 per iteration, one per group) ----
    for (int j = 0; j < S_LEN / 2; ++j) {
        const int s = 2 * j + grp;

        // stage enc[s][b][n0:n0+16][:] into LDS (streamed once from HBM)
        {
            const int tl = tid & 127;
            const v4f* enc4 = (const v4f*)enc;
            v4f* st4 = (v4f*)stage[grp];
            #pragma unroll
            for (int q = 0; q < 2; ++q) {
                int f = tl + 128 * q;
                int r = f >> 4, h4 = f & 15;
                int n = n0 + r; if (n > N_DIM - 1) n = N_DIM - 1;
                size_t g = (size_t)((s * B_DIM + b) * N_DIM + n) * (H_DIM / 4) + h4;
                st4[f] = __builtin_nontemporal_load(&enc4[g]);
            }
        }
        __syncthreads();

        // energy tile: rows = 16 n values, cols = 16*otile..+15, K = 64
        v8f acc0 = {}, acc1 = {};
        const float* st = stage[grp];
        #pragma unroll
        for (int kk = 0; kk < 8; ++kk) {
            v2f a;
            int addr = l15 * H_DIM + 4 * kk + 2 * half;
            a.x = st[addr]; a.y = st[addr + 1];
            acc0 = __builtin_amdgcn_wmma_f32_16x16x4_f32(false, a, false, bf[kk],
                                                         (short)0, acc0, false, false);
        }
        #pragma unroll
        for (int kk = 8; kk < 16; ++kk) {
            v2f a;
            int addr = l15 * H_DIM + 4 * kk + 2 * half;
            a.x = st[addr]; a.y = st[addr + 1];
            acc1 = __builtin_amdgcn_wmma_f32_16x16x4_f32(false, a, false, bf[kk],
                                                         (short)0, acc1, false, false);
        }

        // bias + tanh + dec weighting, reduce over this wave's 16 columns
        float rowsum[8];
        #pragma unroll
        for (int i = 0; i < 8; ++i) {
            int row = i + 8 * half;  // C/D layout: VGPR i -> M = i (+8 for upper half)
            float en = fast_tanh(acc0[i] + acc1[i] + bias_val);
            float w  = en * dec_lds[row * H_DIM + 16 * otile + l15];
            w += __shfl_xor(w, 1, 32);
            w += __shfl_xor(w, 2, 32);
            w += __shfl_xor(w, 4, 32);
            w += __shfl_xor(w, 8, 32);
            rowsum[i] = w;
        }
        if (l15 == 0) {   // lane 0 -> rows 0..7, lane 16 -> rows 8..15
            #pragma unroll
            for (int i = 0; i < 8; ++i)
                red_lds[grp][otile][i + 8 * half] = rowsum[i];
        }
        __syncthreads();

        // deterministic cross-wave combine (wave 0 handles both groups)
        if (tid < 32) {
            int g = tid >> 4, row = tid & 15;
            e_lds[2 * j + g][row] = red_lds[g][0][row] + red_lds[g][1][row]
                                  + red_lds[g][2][row] + red_lds[g][3][row];
        }
        // next iteration's post-stage __syncthreads orders this vs. red_lds reuse
    }
    __syncthreads();

    // ---- softmax over s; write out[b][n][s] ----
    {
        const int n_local = tid >> 4;     // 16 consecutive tids (same half-wave) per n
        const int ls = tid & 15;
        float v0 = e_lds[ls +  0][n_local];
        float v1 = e_lds[ls + 16][n_local];
        float v2 = e_lds[ls + 32][n_local];
        float v3 = e_lds[ls + 48][n_local];
        float m = fmaxf(fmaxf(v0, v1), fmaxf(v2, v3));
        m = fmaxf(m, __shfl_xor(m, 1, 32));
        m = fmaxf(m, __shfl_xor(m, 2, 32));
        m = fmaxf(m, __shfl_xor(m, 4, 32));
        m = fmaxf(m, __shfl_xor(m, 8, 32));
        float e0 = __expf(v0 - m), e1 = __expf(v1 - m);
        float e2 = __expf(v2 - m), e3 = __expf(v3 - m);
        float sum = e0 + e1 + e2 + e3;
        sum += __shfl_xor(sum, 1, 32);
        sum += __shfl_xor(sum, 2, 32);
        sum += __shfl_xor(sum, 4, 32);
        sum += __shfl_xor(sum, 8, 32);
        float inv = 1.f / sum;
        int n = n0 + n_local;
        if (n < N_DIM) {
            float* o = out + (size_t)(b * N_DIM + n) * S_LEN;
            __builtin_nontemporal_store(e0 * inv, &o[ls +  0]);
            __builtin_nontemporal_store(e1 * inv, &o[ls + 16]);
            __builtin_nontemporal_store(e2 * inv, &o[ls + 32]);
            __builtin_nontemporal_store(e3 * inv, &o[ls + 48]);
        }
    }
}

extern "C" void kernel_launch(void* const* d_in, const int* in_sizes, int n_in,
                              void* d_out, int out_size, void* d_ws, size_t ws_size,
                              hipStream_t stream) {
    (void)in_sizes; (void)n_in; (void)out_size; (void)d_ws; (void)ws_size;
    const float* enc  = (const float*)d_in[0];   // (S,B,N,H) fp32
    const float* dec  = (const float*)d_in[1];   // (B,N,H)   fp32
    const float* W    = (const float*)d_in[2];   // (H,H)     fp32
    const float* bias = (const float*)d_in[3];   // (H,)      fp32
    float* out = (float*)d_out;                  // (B,N,S)   fp32

    dim3 grid((N_DIM + TILE_N - 1) / TILE_N, B_DIM);  // 21 x 32 = 672 WGs
    attn_wmma_kernel<<<grid, dim3(256), 0, stream>>>(enc, dec, W, bias, out);
}